// VertexConv_27169963114583
// MI455X (gfx1250) — compile-verified
//
#include <hip/hip_runtime.h>
#include <cstdint>

// ---------------------------------------------------------------------------
// MI455X (gfx1250) fused hypergraph edge-attention.
//   k1: node_scores = feat @ W_att^T                (f32 WMMA 16x16x4)
//   k2: [edge_scores | agg] = inc^T @ [scores|feat] (inc read ONCE, double-
//       buffered TDM->LDS pipeline using TENSORcnt)
//   k3a/k3b: online softmax stats over E axis
//   k4: out = a*prev + (1-a) * (agg*softmax) @ W_proj^T  (WMMA, fused scale+blend)
// ---------------------------------------------------------------------------

typedef __attribute__((ext_vector_type(2))) float v2f;
typedef __attribute__((ext_vector_type(8))) float v8f;
typedef __attribute__((ext_vector_type(4))) unsigned int v4u;
typedef __attribute__((ext_vector_type(8))) int v8i;
typedef __attribute__((ext_vector_type(4))) int v4i;

static constexpr int BB = 4, MM = 8192, EE = 2048, DD = 128;

#define TDM_OK (__has_builtin(__builtin_amdgcn_tensor_load_to_lds) && \
                __has_builtin(__builtin_amdgcn_s_wait_tensorcnt))

__device__ __forceinline__ v8f wmma4(v2f a, v2f b, v8f c) {
  // D = A(16x4 f32) * B(4x16 f32) + C(16x16 f32)
  return __builtin_amdgcn_wmma_f32_16x16x4_f32(false, a, false, b, (short)0, c,
                                               false, false);
}

#if TDM_OK
// Build a 2D Tensor-DMA descriptor (D#) per CDNA5 ISA ch.8 and issue
// tensor_load_to_lds (tile packed contiguously row-by-row into LDS).
__device__ __forceinline__ void tdm_load_2d(unsigned lds_addr, const float* g,
                                            unsigned tile_w, unsigned tile_h,
                                            unsigned tensor_w, unsigned tensor_h,
                                            unsigned row_stride) {
  unsigned long long ga = (unsigned long long)(uintptr_t)g;
  v4u g0 = { 1u,                                   // count=1, user mode
             lds_addr,                             // lds_addr [63:32]
             (unsigned)ga,                         // global_addr[31:0]
             (unsigned)((ga >> 32) & 0x1FFFFFFu) | (2u << 30) };  // addr[56:32] | type=2
  v8i g1 = { (int)(2u << 16),                                     // data_size=4B
             (int)((tensor_w & 0xFFFFu) << 16),                   // tensor_dim0 lo
             (int)((tensor_w >> 16) | ((tensor_h & 0xFFFFu) << 16)),
             (int)((tensor_h >> 16) | (tile_w << 16)),            // tile_dim0
             (int)(tile_h & 0xFFFFu),                             // tile_dim1 (tile_dim2=0)
             (int)row_stride,                                     // dim0 stride lo
             0, 0 };
  v4i gz = {0, 0, 0, 0};
  v8i gz8 = {0, 0, 0, 0, 0, 0, 0, 0};
  __builtin_amdgcn_tensor_load_to_lds(g0, g1, gz, gz, gz8, 0);
}
#endif

// ------------------------- k1: node attention scores -----------------------
__global__ void __launch_bounds__(256) k1_att(const float* __restrict__ feat,
                                              const float* __restrict__ Watt,
                                              float* __restrict__ nsc) {
  const int lane = threadIdx.x & 31, wave = threadIdx.x >> 5;
  const int hl = lane >> 4, l16 = lane & 15;
  const size_t r0 = (size_t)blockIdx.x * 128 + wave * 16;

  const v8f z = {0.f,0.f,0.f,0.f,0.f,0.f,0.f,0.f};
  v8f acc[8];
#pragma unroll
  for (int i = 0; i < 8; ++i) acc[i] = z;

  const float* arow = feat + (r0 + l16) * DD;
#pragma unroll 4
  for (int k0 = 0; k0 < DD; k0 += 4) {
    const int d = k0 + 2 * hl;
    v2f a = { arow[d], arow[d + 1] };
#pragma unroll
    for (int ct = 0; ct < 8; ++ct) {
      const int o = ct * 16 + l16;
      v2f bb = { Watt[o * DD + d], Watt[o * DD + d + 1] };  // L2-resident 64KB
      acc[ct] = wmma4(a, bb, acc[ct]);
    }
  }
#pragma unroll
  for (int ct = 0; ct < 8; ++ct)
#pragma unroll
    for (int r = 0; r < 8; ++r)
      nsc[(r0 + hl * 8 + r) * DD + ct * 16 + l16] = acc[ct][r];
}

// ---------------- k2: fused dual GEMM  inc^T @ [scores | feat] -------------
__global__ void __launch_bounds__(256) k2_dual(const float* __restrict__ inc,
                                               const float* __restrict__ feat,
                                               const float* __restrict__ nsc,
                                               float* __restrict__ esc,
                                               float* __restrict__ agg) {
  // Double-buffered tiles: 2 * (8 + 16 + 16) KB = 80 KB of the 320 KB WGP LDS.
  __shared__ float inc_s[2][32 * 64];
  __shared__ float sc_s[2][32 * 128];
  __shared__ float ft_s[2][32 * 128];

  const int b  = blockIdx.x >> 5;          // 32 e-tiles of 64 per batch
  const int e0 = (blockIdx.x & 31) * 64;
  const int lane = threadIdx.x & 31, wave = threadIdx.x >> 5;
  const int hl = lane >> 4, l16 = lane & 15;
  const int esub = wave & 3;               // which 16-row e sub-tile
  const int cgrp = wave >> 2;              // 0 = scores cols, 1 = feat cols
  const int eloc = esub * 16 + l16;

  const v8f z = {0.f,0.f,0.f,0.f,0.f,0.f,0.f,0.f};
  v8f acc[8];
#pragma unroll
  for (int i = 0; i < 8; ++i) acc[i] = z;

  const int NCH = MM / 32;

#if TDM_OK
  // Prologue: fill buffer 0 with chunk 0.
  if (wave == 0) {
    tdm_load_2d((unsigned)(uintptr_t)inc_s[0], inc + ((size_t)b * MM) * EE + e0,
                64, 32, (unsigned)EE, (unsigned)MM, (unsigned)EE);
    tdm_load_2d((unsigned)(uintptr_t)sc_s[0], nsc + ((size_t)b * MM) * DD,
                128, 32, (unsigned)DD, (unsigned)MM, (unsigned)DD);
    tdm_load_2d((unsigned)(uintptr_t)ft_s[0], feat + ((size_t)b * MM) * DD,
                128, 32, (unsigned)DD, (unsigned)MM, (unsigned)DD);
    __builtin_amdgcn_s_wait_tensorcnt(0);
  }
  __syncthreads();
#endif

  for (int k = 0; k < NCH; ++k) {
    const int buf = k & 1;
    const int m0 = k * 32;
#if TDM_OK
    // Kick off chunk k+1 into the other buffer; overlaps with WMMA below.
    if (wave == 0 && k + 1 < NCH) {
      const int m1 = m0 + 32;
      tdm_load_2d((unsigned)(uintptr_t)inc_s[buf ^ 1],
                  inc + ((size_t)b * MM + m1) * EE + e0,
                  64, 32, (unsigned)EE, (unsigned)MM, (unsigned)EE);
      tdm_load_2d((unsigned)(uintptr_t)sc_s[buf ^ 1],
                  nsc + ((size_t)b * MM + m1) * DD,
                  128, 32, (unsigned)DD, (unsigned)MM, (unsigned)DD);
      tdm_load_2d((unsigned)(uintptr_t)ft_s[buf ^ 1],
                  feat + ((size_t)b * MM + m1) * DD,
                  128, 32, (unsigned)DD, (unsigned)MM, (unsigned)DD);
    }
#else
    {
      const float* incg = inc + ((size_t)b * MM + m0) * EE + e0;
      const float* scg  = nsc + ((size_t)b * MM + m0) * DD;
      const float* ftg  = feat + ((size_t)b * MM + m0) * DD;
      for (int idx = threadIdx.x; idx < 512; idx += 256) {          // 32x64
        int r = idx >> 4, c = (idx & 15) << 2;
        *(float4*)&inc_s[buf][r * 64 + c] = *(const float4*)&incg[(size_t)r * EE + c];
      }
      for (int idx = threadIdx.x; idx < 1024; idx += 256) {         // 32x128 x2
        int r = idx >> 5, c = (idx & 31) << 2;
        *(float4*)&sc_s[buf][r * 128 + c] = *(const float4*)&scg[(size_t)r * DD + c];
        *(float4*)&ft_s[buf][r * 128 + c] = *(const float4*)&ftg[(size_t)r * DD + c];
      }
      __syncthreads();
    }
#endif
    const float* incb = inc_s[buf];
    const float* rhs  = cgrp ? ft_s[buf] : sc_s[buf];
#pragma unroll
    for (int kk = 0; kk < 8; ++kk) {
      const int mloc = kk * 4 + 2 * hl;
      // A = inc^T tile: A[e, m] = inc_s[m][e]
      v2f a = { incb[mloc * 64 + eloc], incb[(mloc + 1) * 64 + eloc] };
#pragma unroll
      for (int ct = 0; ct < 8; ++ct) {
        const int col = ct * 16 + l16;
        v2f bb = { rhs[mloc * 128 + col], rhs[(mloc + 1) * 128 + col] };
        acc[ct] = wmma4(a, bb, acc[ct]);
      }
    }
    __syncthreads();                         // all waves done reading buf
#if TDM_OK
    if (k + 1 < NCH) {
      if (wave == 0) __builtin_amdgcn_s_wait_tensorcnt(0);
      __syncthreads();                       // buf^1 now valid for everyone
    }
#endif
  }

  float* dst = cgrp ? agg : esc;
  const int r0 = e0 + esub * 16 + hl * 8;
#pragma unroll
  for (int ct = 0; ct < 8; ++ct)
#pragma unroll
    for (int r = 0; r < 8; ++r)
      dst[((size_t)b * EE + r0 + r) * DD + ct * 16 + l16] = acc[ct][r];
}

// ----------------- k3a/k3b: online softmax over the E axis -----------------
__global__ void __launch_bounds__(128) k3a_part(const float* __restrict__ esc,
                                                float* __restrict__ pmax,
                                                float* __restrict__ psum) {
  const int b = blockIdx.x >> 4, slab = blockIdx.x & 15, d = threadIdx.x;
  float mx = -3.4e38f, sm = 0.f;
  const int e0 = slab * 128;
  for (int e = e0; e < e0 + 128; ++e) {
    float s = esc[((size_t)b * EE + e) * DD + d];
    if (s > mx) { sm = sm * __expf(mx - s) + 1.f; mx = s; }
    else        { sm += __expf(s - mx); }
  }
  pmax[(b * 16 + slab) * DD + d] = mx;
  psum[(b * 16 + slab) * DD + d] = sm;
}

__global__ void __launch_bounds__(128) k3b_comb(const float* __restrict__ pmax,
                                                const float* __restrict__ psum,
                                                float* __restrict__ gmax,
                                                float* __restrict__ ginv) {
  const int b = blockIdx.x, d = threadIdx.x;
  float mx = -3.4e38f, sm = 0.f;
  for (int s = 0; s < 16; ++s) {
    float m = pmax[(b * 16 + s) * DD + d];
    float v = psum[(b * 16 + s) * DD + d];
    float nm = fmaxf(mx, m);
    sm = sm * __expf(mx - nm) + v * __expf(m - nm);
    mx = nm;
  }
  gmax[b * DD + d] = mx;
  ginv[b * DD + d] = 1.f / sm;
}

// --------- k4: fused attention-scale + projection GEMM + alpha blend -------
__global__ void __launch_bounds__(256) k4_proj(const float* __restrict__ esc,
                                               const float* __restrict__ agg,
                                               const float* __restrict__ gmax,
                                               const float* __restrict__ ginv,
                                               const float* __restrict__ Wp,
                                               const float* __restrict__ prev,
                                               const float* __restrict__ alpha_p,
                                               float* __restrict__ out) {
  __shared__ float smax[DD], sinv[DD];
  const int b  = blockIdx.x >> 4;
  const int e0 = (blockIdx.x & 15) * 128;
  if (threadIdx.x < DD) {
    smax[threadIdx.x] = gmax[b * DD + threadIdx.x];
    sinv[threadIdx.x] = ginv[b * DD + threadIdx.x];
  }
  __syncthreads();

  const int lane = threadIdx.x & 31, wave = threadIdx.x >> 5;
  const int hl = lane >> 4, l16 = lane & 15;
  const int row = e0 + wave * 16 + l16;
  const size_t base = ((size_t)b * EE + row) * DD;

  const v8f z = {0.f,0.f,0.f,0.f,0.f,0.f,0.f,0.f};
  v8f acc[8];
#pragma unroll
  for (int i = 0; i < 8; ++i) acc[i] = z;

#pragma unroll 4
  for (int k0 = 0; k0 < DD; k0 += 4) {
    const int d = k0 + 2 * hl;
    // edge_features = agg * softmax(edge_scores) computed on the fly
    float a0 = agg[base + d]     * __expf(esc[base + d]     - smax[d])     * sinv[d];
    float a1 = agg[base + d + 1] * __expf(esc[base + d + 1] - smax[d + 1]) * sinv[d + 1];
    v2f a = { a0, a1 };
#pragma unroll
    for (int ct = 0; ct < 8; ++ct) {
      const int o = ct * 16 + l16;
      v2f bb = { Wp[o * DD + d], Wp[o * DD + d + 1] };  // L2-resident 64KB
      acc[ct] = wmma4(a, bb, acc[ct]);
    }
  }

  const float al = alpha_p[0];
  const int r0 = e0 + wave * 16 + hl * 8;
#pragma unroll
  for (int ct = 0; ct < 8; ++ct)
#pragma unroll
    for (int r = 0; r < 8; ++r) {
      size_t idx = ((size_t)b * EE + r0 + r) * DD + ct * 16 + l16;
      out[idx] = al * prev[idx] + (1.f - al) * acc[ct][r];
    }
}

// ---------------------------------------------------------------------------
extern "C" void kernel_launch(void* const* d_in, const int* in_sizes, int n_in,
                              void* d_out, int out_size, void* d_ws, size_t ws_size,
                              hipStream_t stream) {
  const float* feat  = (const float*)d_in[0];   // (B,M,D)
  const float* inc   = (const float*)d_in[1];   // (B,M,E)
  const float* prev  = (const float*)d_in[2];   // (B,E,D)
  const float* Watt  = (const float*)d_in[3];   // (D,D)
  const float* Wproj = (const float*)d_in[4];   // (D,D)
  const float* alpha = (const float*)d_in[5];   // scalar
  float* out = (float*)d_out;

  float* ws   = (float*)d_ws;
  float* nsc  = ws;                               // B*M*D
  float* esc  = nsc + (size_t)BB * MM * DD;       // B*E*D
  float* agg  = esc + (size_t)BB * EE * DD;       // B*E*D
  float* pmax = agg + (size_t)BB * EE * DD;       // B*16*D
  float* psum = pmax + (size_t)BB * 16 * DD;      // B*16*D
  float* gmax = psum + (size_t)BB * 16 * DD;      // B*D
  float* ginv = gmax + (size_t)BB * DD;           // B*D

  k1_att<<<BB * MM / 128, 256, 0, stream>>>(feat, Watt, nsc);
  k2_dual<<<BB * (EE / 64), 256, 0, stream>>>(inc, feat, nsc, esc, agg);
  k3a_part<<<BB * 16, 128, 0, stream>>>(esc, pmax, psum);
  k3b_comb<<<BB, 128, 0, stream>>>(pmax, psum, gmax, ginv);
  k4_proj<<<BB * (EE / 128), 256, 0, stream>>>(esc, agg, gmax, ginv, Wproj, prev,
                                               alpha, out);
}